// Encoder_20014547599838
// MI455X (gfx1250) — compile-verified
//
#include <hip/hip_runtime.h>
#include <hip/hip_bf16.h>
#include <math.h>

// Problem constants from the reference
constexpr int NB = 512;   // batch
constexpr int NS = 128;   // sequence / nodes
constexpr int NH = 256;   // hidden
constexpr int NC = 32;    // node encoding channels
constexpr int NL = 56;    // latent

typedef float v2f __attribute__((ext_vector_type(2)));
typedef float v8f __attribute__((ext_vector_type(8)));
typedef int   v4i __attribute__((ext_vector_type(4)));

// gfx1250 async global->LDS path (ASYNCcnt-tracked), guarded so we always
// compile even on toolchains without the builtin.
#if defined(__gfx1250__) && __has_builtin(__builtin_amdgcn_global_load_async_to_lds_b128)
#define USE_ASYNC_LDS 1
typedef __attribute__((address_space(1))) v4i* gptr_v4i;
typedef __attribute__((address_space(3))) v4i* lptr_v4i;
#else
#define USE_ASYNC_LDS 0
#endif

#if USE_ASYNC_LDS && __has_builtin(__builtin_amdgcn_s_wait_asynccnt)
#define WAIT_ASYNC() __builtin_amdgcn_s_wait_asynccnt(0)
#else
#define WAIT_ASYNC() ((void)0)
#endif

// ---------------------------------------------------------------------------
// Generic fp32 WMMA GEMM:  C[M,N] = A[M,K] @ W[N,K]^T + bias[N] + addcol[n*as]
// Block tile: 64(M) x 32(N), 8 waves, each wave owns one 16x16 C tile.
// K staged through DOUBLE-BUFFERED LDS in chunks of 32: async loads for
// chunk i+1 are issued before the WMMA chain of chunk i, so the ASYNCcnt
// wait lands after 8 matrix ops instead of immediately after issue.
// LDS rows padded by 4 floats (stride 36): b128 stores stay 16B aligned and
// fragment-column reads are bank-conflict-free (gcd(36,64)=4).
// Uses V_WMMA_F32_16X16X4_F32 (CDNA5 fp32 matrix pipe).
// ---------------------------------------------------------------------------
#define BM 64
#define BN 32
#define KC 32
#define LPAD 4

__global__ __launch_bounds__(256) void gemm_wmma_f32(
    const float* __restrict__ A, int lda,
    const float* __restrict__ W, int ldw,
    const float* __restrict__ bias,
    const float* __restrict__ addcol, int addstride,
    float* __restrict__ Cout, int ldc,
    int K)
{
  __shared__ float As[2][BM][KC + LPAD];
  __shared__ float Ws[2][BN][KC + LPAD];

  const int tid  = threadIdx.x;
  const int wave = tid >> 5;
  const int lane = tid & 31;
  const int lo   = lane & 15;
  const int hi   = lane >> 4;

  const int m_blk = blockIdx.x * BM;
  const int n_blk = blockIdx.y * BN;
  const int wm    = (wave >> 1) * 16;   // 0,16,32,48
  const int wn    = (wave & 1) * 16;    // 0,16

  // panel-staging map: 16B granules; row = tid/8, col = (tid%8)*4
  const int r0 = tid >> 3;              // 0..31
  const int c0 = (tid & 7) * 4;         // 0,4,...,28

  const float* Ap0 = A + (size_t)(m_blk + r0) * lda + c0;
  const float* Ap1 = A + (size_t)(m_blk + r0 + 32) * lda + c0;
  const float* Wp0 = W + (size_t)(n_blk + r0) * ldw + c0;

  auto issue = [&](int k0, int buf) {
#if USE_ASYNC_LDS
    // GLOBAL_LOAD_ASYNC_TO_LDS_B128: no VGPR round-trip, ASYNCcnt-tracked
    __builtin_amdgcn_global_load_async_to_lds_b128(
        (gptr_v4i)(Ap0 + k0), (lptr_v4i)&As[buf][r0][c0], 0, 0);
    __builtin_amdgcn_global_load_async_to_lds_b128(
        (gptr_v4i)(Ap1 + k0), (lptr_v4i)&As[buf][r0 + 32][c0], 0, 0);
    __builtin_amdgcn_global_load_async_to_lds_b128(
        (gptr_v4i)(Wp0 + k0), (lptr_v4i)&Ws[buf][r0][c0], 0, 0);
#else
    float4 a0 = *(const float4*)(Ap0 + k0);
    float4 a1 = *(const float4*)(Ap1 + k0);
    float4 w0 = *(const float4*)(Wp0 + k0);
    *(float4*)&As[buf][r0][c0]      = a0;
    *(float4*)&As[buf][r0 + 32][c0] = a1;
    *(float4*)&Ws[buf][r0][c0]      = w0;
#endif
  };

  v8f acc = {};
  int buf = 0;

  issue(0, 0);
  WAIT_ASYNC();
  __syncthreads();

  const int nch = K / KC;
  for (int i = 0; i < nch; ++i) {
    if (i + 1 < nch) issue((i + 1) * KC, buf ^ 1);  // prefetch into idle buffer

#pragma unroll
    for (int kk = 0; kk < KC; kk += 4) {
      // A 16x4 fp32 fragment (ISA layout: lanes0-15 K0/K1, lanes16-31 K2/K3)
      v2f a, b;
      a.x = As[buf][wm + lo][kk + 2 * hi];
      a.y = As[buf][wm + lo][kk + 2 * hi + 1];
      // B 4x16 fp32 fragment: row-striped across lanes within a VGPR
      b.x = Ws[buf][wn + lo][kk + hi];
      b.y = Ws[buf][wn + lo][kk + hi + 2];
      acc = __builtin_amdgcn_wmma_f32_16x16x4_f32(
          false, a, false, b, (short)0, acc, false, false);
    }

    WAIT_ASYNC();      // next chunk's LDS writes landed
    __syncthreads();   // (compiler emits s_wait_dscnt 0 first: reads done)
    buf ^= 1;
  }

  const int n = n_blk + wn + lo;
  float add = 0.f;
  if (bias)   add += bias[n];
  if (addcol) add += addcol[(size_t)n * addstride];
#pragma unroll
  for (int r = 0; r < 8; ++r) {
    int m = m_blk + wm + r + 8 * hi;   // ISA C/D layout
    Cout[(size_t)m * ldc + n] = acc[r] + add;
  }
}

// ---------------------------------------------------------------------------
// h_in[b,h] = sum_{j<idx} dep[b,idx,j] * P[b,j,h]   (L2-resident gather)
// ---------------------------------------------------------------------------
__global__ __launch_bounds__(256) void gather_hin(
    const float* __restrict__ dep, const float* __restrict__ P,
    float* __restrict__ h_in, int idx)
{
  const int b = blockIdx.x;
  const int h = threadIdx.x;            // NH == 256 == blockDim
  const float* drow = dep + ((size_t)b * NS + idx) * NS;
  const float* Pb   = P + (size_t)b * NS * NH + h;
  float acc = 0.f;
  for (int j = 0; j < idx; ++j) {
    float m = drow[j];                  // uniform across the block
    if (m != 0.f) acc += m * Pb[(size_t)j * NH];
  }
  h_in[(size_t)b * NH + h] = acc;
}

// ---------------------------------------------------------------------------
// GRU combine: hv = (1-z)*n + z*h_in  (split order r,z,n as in torch GRUCell)
// ---------------------------------------------------------------------------
__global__ __launch_bounds__(256) void gru_combine(
    const float* __restrict__ GI, const float* __restrict__ GH,
    const float* __restrict__ h_in, float* __restrict__ hv)
{
  const int b = blockIdx.x;
  const int h = threadIdx.x;
  const float* gi = GI + (size_t)b * (3 * NH);
  const float* gh = GH + (size_t)b * (3 * NH);
  float r = 1.f / (1.f + __expf(-(gi[h]            + gh[h])));
  float z = 1.f / (1.f + __expf(-(gi[h + NH]       + gh[h + NH])));
  float n = tanhf(gi[h + 2 * NH] + r * gh[h + 2 * NH]);
  float hp = h_in[(size_t)b * NH + h];
  hv[(size_t)b * NH + h] = (1.f - z) * n + z * hp;
}

// ---------------------------------------------------------------------------
// P[b,idx,:] = sigmoid(Gg) * Gm   (per-node product, computed once per node)
// ---------------------------------------------------------------------------
__global__ __launch_bounds__(256) void pstore(
    const float* __restrict__ Gg, const float* __restrict__ Gm,
    float* __restrict__ P, int idx)
{
  const int b = blockIdx.x;
  const int h = threadIdx.x;
  float g = 1.f / (1.f + __expf(-Gg[(size_t)b * NH + h]));
  P[((size_t)b * NS + idx) * NH + h] = g * Gm[(size_t)b * NH + h];
}

// ---------------------------------------------------------------------------
// Emit hv, mu, logvar (tiny: 512x56x256 dots, VALU is fine)
// ---------------------------------------------------------------------------
__global__ __launch_bounds__(256) void finalize_out(
    const float* __restrict__ hv,
    const float* __restrict__ W_mu, const float* __restrict__ b_mu,
    const float* __restrict__ W_lv, const float* __restrict__ b_lv,
    float* __restrict__ out)
{
  const int b = blockIdx.x;
  const int t = threadIdx.x;
  const float* hb = hv + (size_t)b * NH;
  out[(size_t)b * NH + t] = hb[t];
  if (t < NL) {
    float am = b_mu[t], al = b_lv[t];
    for (int k = 0; k < NH; ++k) {
      float x = hb[k];
      am += x * W_mu[t * NH + k];
      al += x * W_lv[t * NH + k];
    }
    out[(size_t)NB * NH + (size_t)b * NL + t] = am;
    out[(size_t)NB * NH + (size_t)NB * NL + (size_t)b * NL + t] = al;
  }
}

// ---------------------------------------------------------------------------
extern "C" void kernel_launch(void* const* d_in, const int* in_sizes, int n_in,
                              void* d_out, int out_size, void* d_ws, size_t ws_size,
                              hipStream_t stream) {
  (void)in_sizes; (void)n_in; (void)out_size; (void)ws_size;
  const float* dep  = (const float*)d_in[0];
  const float* enc  = (const float*)d_in[1];
  const float* Wg   = (const float*)d_in[2];
  const float* bg   = (const float*)d_in[3];
  const float* Wm   = (const float*)d_in[4];
  const float* bm   = (const float*)d_in[5];
  const float* W_ih = (const float*)d_in[6];
  const float* b_ih = (const float*)d_in[7];
  const float* W_hh = (const float*)d_in[8];
  const float* b_hh = (const float*)d_in[9];
  const float* W_mu = (const float*)d_in[10];
  const float* b_mu = (const float*)d_in[11];
  const float* W_lv = (const float*)d_in[12];
  const float* b_lv = (const float*)d_in[13];

  float* ws   = (float*)d_ws;
  float* P    = ws;                               // [NB,NS,NH]  64 MB
  float* h_in = P    + (size_t)NB * NS * NH;      // [NB,NH]
  float* GI   = h_in + (size_t)NB * NH;           // [NB,3H]
  float* GH   = GI   + (size_t)NB * 3 * NH;       // [NB,3H]
  float* hv   = GH   + (size_t)NB * 3 * NH;       // [NB,NH]
  float* Gg   = hv   + (size_t)NB * NH;           // [NB,NH]
  float* Gm   = Gg   + (size_t)NB * NH;           // [NB,NH]

  const dim3 blk(256);
  const dim3 grid_g3h(NB / BM, (3 * NH) / BN);    // (8, 24)
  const dim3 grid_gh (NB / BM, NH / BN);          // (8, 8)

  for (int idx = 0; idx < NS; ++idx) {
    // 1) masked gather of predecessor products (no read-before-write: strictly
    //    lower-triangular dep => only P rows written at earlier steps are read)
    gather_hin<<<NB, blk, 0, stream>>>(dep, P, h_in, idx);

    // 2) GI = enc[:,idx,:] @ W_ih^T + b_ih   [512,768], K=32
    gemm_wmma_f32<<<grid_g3h, blk, 0, stream>>>(
        enc + (size_t)idx * NC, NS * NC, W_ih, NC, b_ih, nullptr, 0,
        GI, 3 * NH, NC);

    // 3) GH = h_in @ W_hh^T + b_hh           [512,768], K=256
    gemm_wmma_f32<<<grid_g3h, blk, 0, stream>>>(
        h_in, NH, W_hh, NH, b_hh, nullptr, 0, GH, 3 * NH, NH);

    // 4) GRU elementwise combine -> hv
    gru_combine<<<NB, blk, 0, stream>>>(GI, GH, h_in, hv);

    // 5) gate/map projections for this node: hv @ Wg[:, :H]^T + Wg[:,H+idx] + bg
    gemm_wmma_f32<<<grid_gh, blk, 0, stream>>>(
        hv, NH, Wg, NH + NS, bg, Wg + NH + idx, NH + NS, Gg, NH, NH);
    gemm_wmma_f32<<<grid_gh, blk, 0, stream>>>(
        hv, NH, Wm, NH + NS, bm, Wm + NH + idx, NH + NS, Gm, NH, NH);

    // 6) P[:,idx,:] = sigmoid(Gg) * Gm
    pstore<<<NB, blk, 0, stream>>>(Gg, Gm, P, idx);
  }

  // hv of last node -> hv, mu, logvar
  finalize_out<<<NB, blk, 0, stream>>>(hv, W_mu, b_mu, W_lv, b_lv, (float*)d_out);
}